// SmartLinearAppearance_68092411510799
// MI455X (gfx1250) — compile-verified
//
#include <hip/hip_runtime.h>

// Problem constants (from reference setup_inputs)
#define BB   8
#define NN   64
#define TT   64
#define DD   1792
#define VV   7
#define TOK  512
#define FF   256          // DD / VV (feature_dim per part)
#define ALPHA_C 0.9f

typedef __attribute__((ext_vector_type(2))) float v2f;
typedef __attribute__((ext_vector_type(4))) float v4f;
typedef __attribute__((ext_vector_type(8))) float v8f;

// ---------------------------------------------------------------------------
// Kernel 1: reversed-time EMA scan. One 448-thread block (14 waves) per (b,n).
// Thread tid owns the 4 consecutive floats d = 4*tid .. 4*tid+3, which all lie
// in part p = tid/64, so each thread tracks exactly one nv/EMA channel and
// issues a single global_load_b128 per timestep (512 B per wave-instruction).
// Streams 235 MB of embs -> pure bandwidth kernel (~10 us at 23.3 TB/s).
// Software double-buffer: the load for t-1 is issued before computing t.
// ---------------------------------------------------------------------------
__global__ __launch_bounds__(448) void smart_scan_kernel(
    const float* __restrict__ embs,   // [B,N,T,D]
    const float* __restrict__ vis,    // [B,N,T,V]
    const int*   __restrict__ masks,  // [B,N,T]
    float* __restrict__ ne_ws,        // [B*N, D]
    float* __restrict__ nm_ws)        // [B*N]
{
  const int bn  = blockIdx.x;       // 0..511
  const int tid = threadIdx.x;      // 0..447
  const int p   = tid >> 6;         // part / vis channel: (4*tid)/256

  const float* e_base = embs  + (size_t)bn * TT * DD + 4 * tid;
  const float* v_base = vis   + (size_t)bn * TT * VV + p;
  const int*   m_base = masks + (size_t)bn * TT;

  v4f   ne4 = {};
  float nv  = 0.0f;
  int   anym = 0;

  // preload t = TT-1
  v4f e = *(const v4f*)(e_base + (size_t)(TT - 1) * DD);

  for (int t = TT - 1; t >= 0; --t) {
    // issue next timestep's load early (in flight behind this step's compute)
    v4f e_next = {};
    if (t > 0) e_next = *(const v4f*)(e_base + (size_t)(t - 1) * DD);

    const int   m = m_base[t];            // uniform across block
    const float v = v_base[(size_t)t * VV];

    const float x    = ((nv != 0.0f) != (v != 0.0f)) ? 1.0f : 0.0f;
    const float prod = nv * v;
    const float ema_tr  = prod * ALPHA_C          + x * nv;
    const float ema_det = prod * (1.0f - ALPHA_C) + x * v;

    if (m) {
      ne4.x = ema_tr * ne4.x + ema_det * e.x;
      ne4.y = ema_tr * ne4.y + ema_det * e.y;
      ne4.z = ema_tr * ne4.z + ema_det * e.z;
      ne4.w = ema_tr * ne4.w + ema_det * e.w;
      nv = fmaxf(nv, v);
    }
    anym |= m;
    e = e_next;
  }

  *(v4f*)(ne_ws + (size_t)bn * DD + 4 * tid) = ne4;
  if (tid == 0) nm_ws[bn] = anym ? 1.0f : 0.0f;
}

// ---------------------------------------------------------------------------
// Kernel 2: tok[bn, t] = sum_d ne[bn,d] * W[t,d] + b[t], masked by nm[bn].
// FP32 WMMA (V_WMMA_F32_16X16X4_F32). Each wave computes a 16x32 output strip:
// one A-fragment shared by two B-fragments -> two independent WMMA chains and
// half the A-side L2 traffic. 512 waves cover the 512x512 output.
//
// A-frag (16x4 f32): lanes 0-15 -> M=l, K={0,1}; lanes 16-31 -> M=l, K={2,3}
// B-frag (4x16 f32): lanes 0-15 -> N=l, K={0,1}; lanes 16-31 -> N=l, K={2,3}
// C/D (16x16 f32):  VGPR r -> M=r (lanes 0-15) / M=r+8 (lanes 16-31), N=l
// ---------------------------------------------------------------------------
__global__ __launch_bounds__(128) void smart_gemm_wmma_kernel(
    const float* __restrict__ ne_ws,  // [512, 1792]  (A, row-major)
    const float* __restrict__ nm_ws,  // [512]
    const float* __restrict__ Wm,     // [512, 1792]  (tokens x D, row-major)
    const float* __restrict__ bias,   // [512]
    float* __restrict__ out)          // [512, 512]
{
  const int wave = threadIdx.x >> 5;          // 0..3 (4 waves / block)
  const int lane = threadIdx.x & 31;
  const int strip = blockIdx.x * 4 + wave;    // 0..511
  const int tm    = strip >> 4;               // row tile   0..31 (16 rows)
  const int tn2   = strip & 15;               // col strip  0..15 (32 cols)

  const int half = lane >> 4;                 // 0: K={0,1}, 1: K={2,3}
  const int l    = lane & 15;

  const float* a_ptr  = ne_ws + (size_t)(tm  * 16 + l) * DD + 2 * half;
  const float* b0_ptr = Wm    + (size_t)(tn2 * 32 + l) * DD + 2 * half;
  const float* b1_ptr = b0_ptr + (size_t)16 * DD;

  v8f c0 = {};
  v8f c1 = {};
#pragma unroll 4
  for (int k0 = 0; k0 < DD; k0 += 4) {
    v2f a  = *(const v2f*)(a_ptr  + k0);
    v2f b0 = *(const v2f*)(b0_ptr + k0);
    v2f b1 = *(const v2f*)(b1_ptr + k0);
    // (neg_a, A, neg_b, B, c_mod, C, reuse_a, reuse_b)
    c0 = __builtin_amdgcn_wmma_f32_16x16x4_f32(false, a, false, b0,
                                               (short)0, c0, false, false);
    c1 = __builtin_amdgcn_wmma_f32_16x16x4_f32(false, a, false, b1,
                                               (short)0, c1, false, false);
  }

  const int   col0 = tn2 * 32 + l;
  const int   col1 = col0 + 16;
  const float bc0  = bias[col0];
  const float bc1  = bias[col1];
#pragma unroll
  for (int r = 0; r < 8; ++r) {
    const int   row = tm * 16 + r + half * 8;
    const float nm  = nm_ws[row];
    const bool  on  = (nm != 0.0f);
    out[(size_t)row * TOK + col0] = on ? (c0[r] + bc0) : 0.0f;
    out[(size_t)row * TOK + col1] = on ? (c1[r] + bc1) : 0.0f;
  }
}

// ---------------------------------------------------------------------------
extern "C" void kernel_launch(void* const* d_in, const int* in_sizes, int n_in,
                              void* d_out, int out_size, void* d_ws, size_t ws_size,
                              hipStream_t stream) {
  const float* embs  = (const float*)d_in[0];
  const float* vis   = (const float*)d_in[1];
  const int*   masks = (const int*)  d_in[2];
  const float* Wm    = (const float*)d_in[3];
  const float* bias  = (const float*)d_in[4];
  float* out = (float*)d_out;

  float* ne_ws = (float*)d_ws;                                   // 512*1792*4 B
  float* nm_ws = (float*)((char*)d_ws +
                          (size_t)BB * NN * DD * sizeof(float)); // + 512*4 B

  smart_scan_kernel<<<BB * NN, 448, 0, stream>>>(embs, vis, masks, ne_ws, nm_ws);
  smart_gemm_wmma_kernel<<<(32 * 16) / 4, 128, 0, stream>>>(ne_ws, nm_ws, Wm,
                                                            bias, out);
}